// CondCondKernel_46420006535561
// MI455X (gfx1250) — compile-verified
//
#include <hip/hip_runtime.h>
#include <math.h>

typedef __attribute__((ext_vector_type(2))) float v2f;
typedef __attribute__((ext_vector_type(8))) float v8f;

namespace gmm {

constexpr int Dd = 64;   // feature / gaussian dim
constexpr int Kk = 32;   // mixture components

// ---------------- deterministic bucketing by head ----------------

__global__ void hist_kernel(const int* __restrict__ y, int B, int* __restrict__ cc) {
  __shared__ int cnt[4];
  int t = threadIdx.x;
  if (t < 4) cnt[t] = 0;
  __syncthreads();
  int g = blockIdx.x * 256 + t;
  if (g < B) atomicAdd(&cnt[y[g] & 3], 1);
  __syncthreads();
  if (t < 4) cc[blockIdx.x * 4 + t] = cnt[t];
}

__global__ void scan_kernel(const int* __restrict__ cc, int nch,
                            int* __restrict__ cb, int* __restrict__ ps,
                            int* __restrict__ idx) {
  // single thread: tiny (4 heads x nch chunks)
  int total[4] = {0, 0, 0, 0};
  for (int c = 0; c < nch; c++)
    for (int h = 0; h < 4; h++) total[h] += cc[c * 4 + h];
  int start = 0;
  for (int h = 0; h < 4; h++) {
    ps[h] = start;
    int padded = (total[h] + 31) & ~31;
    for (int i = start + total[h]; i < start + padded; i++) idx[i] = -1;  // pad slots
    start += padded;
  }
  ps[4] = start;
  int run[4] = {ps[0], ps[1], ps[2], ps[3]};
  for (int c = 0; c < nch; c++)
    for (int h = 0; h < 4; h++) { cb[c * 4 + h] = run[h]; run[h] += cc[c * 4 + h]; }
}

__global__ void scatter_kernel(const int* __restrict__ y, int B,
                               const int* __restrict__ cb, int* __restrict__ idx) {
  __shared__ int wcnt[8][4];
  __shared__ int wbase[8][4];
  int t = threadIdx.x, lane = t & 31, wv = t >> 5;
  int g = blockIdx.x * 256 + t;
  int hv = (g < B) ? (y[g] & 3) : -1;
  int rank = 0;
  unsigned int ltmask = (lane == 31) ? 0x7fffffffu : ((1u << lane) - 1u);
  for (int h = 0; h < 4; h++) {
    unsigned int m = __builtin_amdgcn_ballot_w32(hv == h);
    if (lane == 0) wcnt[wv][h] = __popc(m);
    if (hv == h) rank = __popc(m & ltmask);
  }
  __syncthreads();
  if (t == 0) {
    int run[4] = {0, 0, 0, 0};
    for (int w = 0; w < 8; w++)
      for (int h = 0; h < 4; h++) { wbase[w][h] = run[h]; run[h] += wcnt[w][h]; }
  }
  __syncthreads();
  if (g < B) idx[cb[blockIdx.x * 4 + hv] + wbase[wv][hv] + rank] = g;
}

// ---------------- fused per-tile kernel ----------------

struct P {
  const float* x; const float* zd;
  const float* W1[3]; const float* b1[3];   // modules: 0=mu 1=lv 2=w
  const float* W2[3]; const float* b2[3];
  const float* muW3; const float* mub3;
  const float* lvW3; const float* lvb3;
  const float* wW3;  const float* wb3;
  const int* idx; const int* ps;
  float* partials;
};

// C(16x16,f32) += A(16x64 rows from LDS) x W(64xldw global, col block)
// fp32 WMMA: A frag = {A[row][k0+2h], A[row][k0+2h+1]}, B frag mirrored on lane=N.
__device__ __forceinline__ v8f wmma_k64(const float* A, int arow,
                                        const float* W, int ldw,
                                        int col, int half) {
  v8f acc = {0.f, 0.f, 0.f, 0.f, 0.f, 0.f, 0.f, 0.f};
  const float* ap = A + arow * Dd + 2 * half;
  const float* wp = W + (size_t)(2 * half) * ldw + col;
  #pragma unroll
  for (int k0 = 0; k0 < Dd; k0 += 4) {
    v2f a; a.x = ap[0]; a.y = ap[1];
    v2f b; b.x = wp[0]; b.y = wp[ldw];
    ap += 4; wp += (size_t)4 * ldw;
    acc = __builtin_amdgcn_wmma_f32_16x16x4_f32(false, a, false, b,
                                                (short)0, acc, false, false);
  }
  return acc;
}

__global__ __launch_bounds__(256) void gmm_fused_kernel(P p) {
  __shared__ float xt[32 * Dd];        // relu(x) tile
  __shared__ float zds[32 * Dd];       // z_d tile
  __shared__ float hbuf[32 * Dd];      // layer ping buffer
  __shared__ float h2[3][32 * Dd];     // h2 per module
  __shared__ float wout[32 * Kk];      // raw w-head logits
  __shared__ float zmat[32 * Kk];      // core + logit per comp
  __shared__ float zred[4][32];        // cross-wave partials
  __shared__ int sidx[32];
  __shared__ int shh;

  int tid = threadIdx.x;
  int tile = blockIdx.x;
  int slot0 = tile * 32;
  if (slot0 >= p.ps[4]) { if (tid == 0) p.partials[tile] = 0.f; return; }
  if (tid == 0) {
    int h = 0;
    while (h < 3 && slot0 >= p.ps[h + 1]) h++;
    shh = h;
  }
  if (tid < 32) sidx[tid] = p.idx[slot0 + tid];
  __syncthreads();
  int h = shh;

  {  // stage inputs (relu applied to x)
    int row = tid >> 3;
    int c0 = (tid & 7) * 8;
    int b = sidx[row];
    const float* xr = p.x  + (size_t)(b < 0 ? 0 : b) * Dd;
    const float* zr = p.zd + (size_t)(b < 0 ? 0 : b) * Dd;
    #pragma unroll
    for (int j = 0; j < 8; j++) {
      float xv = (b >= 0) ? xr[c0 + j] : 0.f;
      xt[row * Dd + c0 + j]  = fmaxf(xv, 0.f);
      zds[row * Dd + c0 + j] = (b >= 0) ? zr[c0 + j] : 0.f;
    }
  }
  __syncthreads();

  int lane = tid & 31, wv = tid >> 5;
  int rt = wv >> 2, nc = wv & 3;        // row tile (0/1), col chunk (0..3)
  int half = lane >> 4, l15 = lane & 15;
  int colg = nc * 16 + l15;             // 0..63
  int arow = rt * 16 + l15;

  // three MLP trunks (layers 1+2)
  #pragma unroll
  for (int m = 0; m < 3; m++) {
    v8f a1 = wmma_k64(xt, arow, p.W1[m] + h * Dd * Dd, Dd, colg, half);
    float bia = p.b1[m][h * Dd + colg];
    __syncthreads();                       // prior hbuf readers done
    #pragma unroll
    for (int r = 0; r < 8; r++)
      hbuf[(rt * 16 + r + 8 * half) * Dd + colg] = fmaxf(a1[r] + bia, 0.f);
    __syncthreads();
    v8f a2 = wmma_k64(hbuf, arow, p.W2[m] + h * Dd * Dd, Dd, colg, half);
    float bib = p.b2[m][h * Dd + colg];
    #pragma unroll
    for (int r = 0; r < 8; r++)
      h2[m][(rt * 16 + r + 8 * half) * Dd + colg] = fmaxf(a2[r] + bib, 0.f);
  }
  __syncthreads();

  // w-head output layer: 64 -> 32 (waves nc<2 only; wave-uniform branch)
  if (nc < 2) {
    int colw = nc * 16 + l15;
    v8f aw = wmma_k64(h2[2], arow, p.wW3 + h * Dd * Kk, Kk, colw, half);
    float biw = p.wb3[h * Kk + colw];
    #pragma unroll
    for (int r = 0; r < 8; r++)
      wout[(rt * 16 + r + 8 * half) * Kk + colw] = aw[r] + biw;
  }
  __syncthreads();

  float zdr[8];
  #pragma unroll
  for (int r = 0; r < 8; r++)
    zdr[r] = zds[(rt * 16 + r + 8 * half) * Dd + colg];

  const float* Wm = p.muW3 + (size_t)h * Dd * (Kk * Dd);
  const float* Wl = p.lvW3 + (size_t)h * Dd * (Kk * Dd);
  const float* Bm = p.mub3 + (size_t)h * (Kk * Dd);
  const float* Bl = p.lvb3 + (size_t)h * (Kk * Dd);

  // mixture-component loop: mu_k / logvar_k tiles via WMMA, fold immediately
  for (int kc = 0; kc < Kk; kc++) {
    v8f am = wmma_k64(h2[0], arow, Wm + kc * Dd, Kk * Dd, colg, half);
    v8f al = wmma_k64(h2[1], arow, Wl + kc * Dd, Kk * Dd, colg, half);
    float bm = Bm[kc * Dd + colg];
    float bl = Bl[kc * Dd + colg];
    float vs[8];
    #pragma unroll
    for (int r = 0; r < 8; r++) {
      float mu  = am[r] + bm;
      float lv  = tanhf(al[r] + bl);
      float var = __expf(lv);
      float t   = (zdr[r] - mu) * var;
      float c   = __logf(var + 1e-8f) - 0.5f * t * t;
      // sum over the 16 d-columns held by this lane-half (deterministic order)
      c += __shfl_xor(c, 1, 32);
      c += __shfl_xor(c, 2, 32);
      c += __shfl_xor(c, 4, 32);
      c += __shfl_xor(c, 8, 32);
      vs[r] = c;
    }
    if (l15 == 0) {
      #pragma unroll
      for (int r = 0; r < 8; r++)
        zred[nc][rt * 16 + r + 8 * half] = vs[r];
    }
    __syncthreads();
    if (wv == 0) {
      int row = lane;
      zmat[row * Kk + kc] = zred[0][row] + zred[1][row] + zred[2][row]
                          + zred[3][row] + wout[row * Kk + kc];
    }
    __syncthreads();
  }

  // per-row: logpdf = logC + lse(core + wlogit) - lse(wlogit)
  if (wv == 0) {
    int row = lane;
    float m1 = -__builtin_inff(), m2 = -__builtin_inff();
    for (int kc = 0; kc < Kk; kc++) {
      m1 = fmaxf(m1, zmat[row * Kk + kc]);
      m2 = fmaxf(m2, wout[row * Kk + kc]);
    }
    float s1 = 0.f, s2 = 0.f;
    for (int kc = 0; kc < Kk; kc++) {
      s1 += __expf(zmat[row * Kk + kc] - m1);
      s2 += __expf(wout[row * Kk + kc] - m2);
    }
    float lse1 = m1 + __logf(s1);
    float lse2 = m2 + __logf(s2);
    const float logC = -32.f * 1.8378770664093453f;  // -D/2 * log(2*pi)
    float val = (sidx[row] >= 0) ? -(logC + lse1 - lse2) : 0.f;
    val += __shfl_xor(val, 1, 32);
    val += __shfl_xor(val, 2, 32);
    val += __shfl_xor(val, 4, 32);
    val += __shfl_xor(val, 8, 32);
    val += __shfl_xor(val, 16, 32);
    if (lane == 0) p.partials[tile] = val;
  }
}

__global__ void reduce_kernel(const float* __restrict__ partials, int n, int B,
                              float* __restrict__ out) {
  if (threadIdx.x == 0 && blockIdx.x == 0) {
    float s = 0.f;
    for (int i = 0; i < n; i++) s += partials[i];  // fixed order: deterministic
    out[0] = s / (float)B;
  }
}

}  // namespace gmm

extern "C" void kernel_launch(void* const* d_in, const int* in_sizes, int n_in,
                              void* d_out, int out_size, void* d_ws, size_t ws_size,
                              hipStream_t stream) {
  const float* x    = (const float*)d_in[0];
  const float* zd   = (const float*)d_in[1];
  const int*   y    = (const int*)d_in[2];
  const float* muW1 = (const float*)d_in[3];
  const float* mub1 = (const float*)d_in[4];
  const float* muW2 = (const float*)d_in[5];
  const float* mub2 = (const float*)d_in[6];
  const float* muW3 = (const float*)d_in[7];
  const float* mub3 = (const float*)d_in[8];
  const float* lvW1 = (const float*)d_in[9];
  const float* lvb1 = (const float*)d_in[10];
  const float* lvW2 = (const float*)d_in[11];
  const float* lvb2 = (const float*)d_in[12];
  const float* lvW3 = (const float*)d_in[13];
  const float* lvb3 = (const float*)d_in[14];
  const float* wW1  = (const float*)d_in[15];
  const float* wb1  = (const float*)d_in[16];
  const float* wW2  = (const float*)d_in[17];
  const float* wb2  = (const float*)d_in[18];
  const float* wW3  = (const float*)d_in[19];
  const float* wb3  = (const float*)d_in[20];

  int B = in_sizes[0] / 64;
  int nch = (B + 255) / 256;
  int maxTiles = (B + 31) / 32 + 4;

  char* w = (char*)d_ws;
  int* idx = (int*)w;           w += sizeof(int) * (size_t)(B + 128);
  int* cc  = (int*)w;           w += sizeof(int) * (size_t)(nch * 4);
  int* cb  = (int*)w;           w += sizeof(int) * (size_t)(nch * 4);
  int* ps  = (int*)w;           w += sizeof(int) * 8;
  float* partials = (float*)w;  w += sizeof(float) * (size_t)maxTiles;
  (void)ws_size; (void)n_in; (void)out_size;

  gmm::hist_kernel<<<nch, 256, 0, stream>>>(y, B, cc);
  gmm::scan_kernel<<<1, 1, 0, stream>>>(cc, nch, cb, ps, idx);
  gmm::scatter_kernel<<<nch, 256, 0, stream>>>(y, B, cb, idx);

  gmm::P p;
  p.x = x; p.zd = zd;
  p.W1[0] = muW1; p.W1[1] = lvW1; p.W1[2] = wW1;
  p.b1[0] = mub1; p.b1[1] = lvb1; p.b1[2] = wb1;
  p.W2[0] = muW2; p.W2[1] = lvW2; p.W2[2] = wW2;
  p.b2[0] = mub2; p.b2[1] = lvb2; p.b2[2] = wb2;
  p.muW3 = muW3; p.mub3 = mub3;
  p.lvW3 = lvW3; p.lvb3 = lvb3;
  p.wW3  = wW3;  p.wb3  = wb3;
  p.idx = idx; p.ps = ps; p.partials = partials;

  gmm::gmm_fused_kernel<<<maxTiles, 256, 0, stream>>>(p);
  gmm::reduce_kernel<<<1, 1, 0, stream>>>(partials, maxTiles, B, (float*)d_out);
}